// TripletSoftmaxLoss_71133248356681
// MI455X (gfx1250) — compile-verified
//
#include <hip/hip_runtime.h>
#include <cstdint>
#include <math.h>

// ---------------------------------------------------------------------------
// TripletSoftmaxLoss for MI455X (gfx1250).
//   K0: global ~3% quantile threshold from 2048 samples (bitwise radix search)
//   K1: one block per row: TDM row -> LDS, keys -> regs, compact survivors,
//       exact top-128 by all-pairs rank; exact radix-search fallback.
//   K2: deterministic tree mean.
// ---------------------------------------------------------------------------

#define NMAT   8192
#define TPB    256
#define EPT    32          // elements per thread (NMAT / TPB)
#define VEC    8           // float4 chunks per thread
#define CAP    1024        // survivor capacity
#define KTOP   128         // n_neg
#define SRANK  61          // ~3% of 2048 samples

typedef unsigned int uint;
typedef uint u32x4 __attribute__((ext_vector_type(4)));
typedef int  i32x4 __attribute__((ext_vector_type(4)));
typedef int  i32x8 __attribute__((ext_vector_type(8)));

#if defined(__gfx1250__) && __has_builtin(__builtin_amdgcn_tensor_load_to_lds)
#define USE_TDM 1
#endif

// Monotone map: float -> uint so that unsigned compare == float compare.
__device__ __forceinline__ uint f2key(float f) {
    uint b = __float_as_uint(f);
    uint m = (uint)((int)b >> 31);
    return b ^ (m | 0x80000000u);
}
__device__ __forceinline__ float key2f(uint k) {
    uint b = (k & 0x80000000u) ? (k ^ 0x80000000u) : ~k;
    return __uint_as_float(b);
}

// ---------------------------------------------------------------------------
// K0: global threshold ~ top-3% quantile from 2048 strided samples.
// ---------------------------------------------------------------------------
__global__ void tsl_threshold(const float* __restrict__ s, uint* __restrict__ wsT) {
    const int tid = threadIdx.x;
    uint kv[8];
#pragma unroll
    for (int q = 0; q < 8; ++q) {
        int j   = tid * 8 + q;
        int row = j * 4;                         // rows 0,4,...,8188
        int col = (j * 2053 + 7) & (NMAT - 1);   // pseudo-random column
        if (col == row) col = (col + 1) & (NMAT - 1);  // never sample diagonal
        kv[q] = f2key(s[(size_t)row * NMAT + col]);
    }
    __shared__ int sc;
    uint prefix = 0;
    for (int b = 31; b >= 0; --b) {
        uint cand = prefix | (1u << b);
        int c = 0;
#pragma unroll
        for (int q = 0; q < 8; ++q) c += (kv[q] >= cand) ? 1 : 0;
        if (tid == 0) sc = 0;
        __syncthreads();
        atomicAdd(&sc, c);
        __syncthreads();
        int C = sc;
        __syncthreads();
        if (C >= SRANK) prefix = cand;
    }
    if (tid == 0) wsT[0] = prefix;
}

// ---------------------------------------------------------------------------
// K1: one block per row.
// ---------------------------------------------------------------------------
__global__ void __launch_bounds__(TPB) tsl_rows(const float* __restrict__ s,
                                                const uint*  __restrict__ wsT,
                                                float* __restrict__ losses) {
    const int row = blockIdx.x;
    const int tid = threadIdx.x;
    const float* srow = s + (size_t)row * NMAT;

    __shared__ int   s_m;
    __shared__ float sPos;
    __shared__ float redf[TPB];
    __shared__ __align__(16) uint skeys[CAP + 4];
#if defined(USE_TDM)
    __shared__ __align__(16) float lrow[NMAT];
#endif

    const uint T = wsT[0];

#if defined(USE_TDM)
    // Wave 0 issues a single Tensor-DMA for the whole 32KB row (EXEC ignored;
    // guard so only one wave issues it), then fences on TENSORcnt.
    if (tid < 32) {
        unsigned long long ga = (unsigned long long)(uintptr_t)srow;
        uint ldsoff = (uint)(uintptr_t)(&lrow[0]);  // low 32b of generic = LDS offset
        u32x4 g0 = { 1u,                                   // count=1 (valid D#)
                     ldsoff,                               // lds_addr
                     (uint)(ga & 0xFFFFFFFFull),           // global_addr[31:0]
                     (((uint)(ga >> 32)) & 0x01FFFFFFu) | 0x80000000u }; // addr[56:32] | type=2
        i32x8 g1 = { 0, 0, 0, 0, 0, 0, 0, 0 };
        g1[0] = 0x00020000;        // workgroup_mask=0, data_size=2 (4 bytes)
        g1[1] = (int)0x20000000;   // tensor_dim0 = 8192 (bits 63:48)
        g1[2] = 0x00010000;        // tensor_dim1 = 1    (bits 95:80)
        g1[3] = (int)0x20000000;   // tile_dim0   = 8192 (bits 127:112)
        g1[4] = 1;                 // tile_dim1   = 1
        g1[5] = NMAT;              // tensor_dim0_stride = 8192
        i32x4 gz = { 0, 0, 0, 0 };
#if __clang_major__ >= 23
        i32x8 z8 = { 0, 0, 0, 0, 0, 0, 0, 0 };
        __builtin_amdgcn_tensor_load_to_lds(g0, g1, gz, gz, z8, 0);
#else
        __builtin_amdgcn_tensor_load_to_lds(g0, g1, gz, gz, 0);
#endif
#if __has_builtin(__builtin_amdgcn_s_wait_tensorcnt)
        __builtin_amdgcn_s_wait_tensorcnt(0);
#else
        asm volatile("s_wait_tensorcnt 0" ::: "memory");
#endif
    }
    __syncthreads();
#endif

    // Pull 32 elements/thread into registers as orderable keys.
    uint kv[EPT];
    float fPos   = 0.0f;
    bool  hasPos = false;
#pragma unroll
    for (int v = 0; v < VEC; ++v) {
        int e0 = (v * TPB + tid) * 4;   // bank-conflict-free, coalesced layout
#if defined(USE_TDM)
        float4 f = *(const float4*)&lrow[e0];
#else
        float4 f = ((const float4*)srow)[v * TPB + tid];
#endif
        float vals[4] = { f.x, f.y, f.z, f.w };
#pragma unroll
        for (int c = 0; c < 4; ++c) {
            int  e = e0 + c;
            uint k = f2key(vals[c]);
            if (e == row) { fPos = vals[c]; hasPos = true; k = 0u; }  // exclude diagonal
            kv[v * 4 + c] = k;
        }
    }
    if (hasPos)   sPos = fPos;
    if (tid == 0) s_m  = 0;
    __syncthreads();

    // Compact survivors (keys >= T) into LDS.
    int c = 0;
#pragma unroll
    for (int j = 0; j < EPT; ++j) c += (kv[j] >= T) ? 1 : 0;
    int base = atomicAdd(&s_m, c);
#pragma unroll
    for (int j = 0; j < EPT; ++j) {
        if (kv[j] >= T) {
            if (base < CAP) skeys[base] = kv[j];
            base++;
        }
    }
    __syncthreads();
    const int M = s_m;   // uniform across block

    if (M >= KTOP && M <= CAP) {
        // Exact top-128 among M survivors: all-pairs rank, (key, index) strict order.
        int Mpad = (M + 3) & ~3;
        if (tid < Mpad - M) skeys[M + tid] = 0u;   // pad (key 0 never wins)
        __syncthreads();
        float acc = 0.0f;
        for (int i = tid; i < M; i += TPB) {
            uint ki  = skeys[i];
            int  cnt = 0;
            for (int j = 0; j < Mpad; j += 4) {
                uint4 q = *(const uint4*)&skeys[j];   // wave-broadcast LDS reads
                cnt += (q.x > ki) + (q.y > ki) + (q.z > ki) + (q.w > ki);
                cnt += ((q.x == ki) & (j + 0 < i)) + ((q.y == ki) & (j + 1 < i))
                     + ((q.z == ki) & (j + 2 < i)) + ((q.w == ki) & (j + 3 < i));
            }
            if (cnt < KTOP) acc += expf(key2f(ki));   // rank < 128 -> in top-128
        }
        redf[tid] = acc;
    } else {
        // Exact fallback (rare / adversarial): bitwise radix search for the
        // 128th-largest key over register-resident row, then tie-corrected sum.
        __shared__ int sc2;
        uint prefix = 0;
        for (int b = 31; b >= 0; --b) {
            uint cand = prefix | (1u << b);
            int cc = 0;
#pragma unroll
            for (int j = 0; j < EPT; ++j) cc += (kv[j] >= cand) ? 1 : 0;
            if (tid == 0) sc2 = 0;
            __syncthreads();
            atomicAdd(&sc2, cc);
            __syncthreads();
            int C = sc2;
            __syncthreads();
            if (C >= KTOP) prefix = cand;
        }
        int cg = 0; float acc = 0.0f;
#pragma unroll
        for (int j = 0; j < EPT; ++j) {
            if (kv[j] > prefix) { cg++; acc += expf(key2f(kv[j])); }
        }
        if (tid == 0) sc2 = 0;
        __syncthreads();
        atomicAdd(&sc2, cg);
        __syncthreads();
        int G = sc2;   // strictly-greater count, guaranteed < 128
        if (tid == 0) acc += (float)(KTOP - G) * expf(key2f(prefix));
        redf[tid] = acc;
    }
    __syncthreads();

    // Deterministic tree reduction of the negative-sum.
#pragma unroll
    for (int st = TPB / 2; st > 0; st >>= 1) {
        if (tid < st) redf[tid] += redf[tid + st];
        __syncthreads();
    }
    if (tid == 0) {
        float pos = expf(sPos);
        losses[row] = log1pf(redf[0] / pos);   // == -log(pos/(pos+negsum))
    }
}

// ---------------------------------------------------------------------------
// K2: deterministic mean of per-row losses.
// ---------------------------------------------------------------------------
__global__ void tsl_mean(const float* __restrict__ losses, float* __restrict__ out) {
    __shared__ float redf[TPB];
    const int tid = threadIdx.x;
    float a = 0.0f;
    for (int k = tid; k < NMAT; k += TPB) a += losses[k];   // fixed order
    redf[tid] = a;
    __syncthreads();
#pragma unroll
    for (int st = TPB / 2; st > 0; st >>= 1) {
        if (tid < st) redf[tid] += redf[tid + st];
        __syncthreads();
    }
    if (tid == 0) out[0] = redf[0] / (float)NMAT;
}

extern "C" void kernel_launch(void* const* d_in, const int* in_sizes, int n_in,
                              void* d_out, int out_size, void* d_ws, size_t ws_size,
                              hipStream_t stream) {
    (void)in_sizes; (void)n_in; (void)out_size; (void)ws_size;
    const float* s      = (const float*)d_in[0];
    uint*        wsT    = (uint*)d_ws;
    float*       losses = (float*)d_ws + 64;   // losses at 256B offset
    float*       out    = (float*)d_out;

    tsl_threshold<<<1,    TPB, 0, stream>>>(s, wsT);
    tsl_rows     <<<NMAT, TPB, 0, stream>>>(s, wsT, losses);
    tsl_mean     <<<1,    TPB, 0, stream>>>(losses, out);
}